// VectorQuantizerEMA_unshared_codebook_49838800503002
// MI455X (gfx1250) — compile-verified
//
#include <hip/hip_runtime.h>

// Problem constants (from reference)
#define B_ 2048
#define D_ 128
#define K_ 512
#define BT 32   // batches per block (one per lane of a wave)
#define DT 8    // dims per block (one per wave; 8 waves = 256 threads)

typedef unsigned int u32;
typedef unsigned long long u64;
typedef u32   v4u __attribute__((ext_vector_type(4)));
typedef int   v8i __attribute__((ext_vector_type(8)));
typedef int   v4i __attribute__((ext_vector_type(4)));
typedef float v2f __attribute__((ext_vector_type(2)));
typedef float v8f __attribute__((ext_vector_type(8)));

// ---- Tensor DMA Descriptor (D#) builders, per CDNA5 ISA §8.3 / §8.4 ----
// Group 0 (128b): [1:0]=count=1, [2]=is_restore=0, [31:30]=gather off,
//                 [63:32]=lds_addr, [120:64]=global_addr, [127:126]=type=2
__device__ __forceinline__ v4u tdm_g0(u32 lds_addr, u64 gaddr) {
    v4u g0;
    g0.x = 1u;                                            // count = 1 (valid)
    g0.y = lds_addr;                                      // LDS byte address
    g0.z = (u32)(gaddr & 0xFFFFFFFFull);                  // global_addr[31:0]
    g0.w = (u32)((gaddr >> 32) & 0x01FFFFFFull)           // global_addr[56:32]
         | (2u << 30);                                    // type = 2 ("image")
    return g0;
}

// Group 1 (256b): wg_mask=0, data_size=2 (4B), no barrier/iterate/pad,
//   tensor_dim0/1, tile_dim0/1 (tile_dim2=0), tensor_dim0_stride, dim1_stride=0
__device__ __forceinline__ v8i tdm_g1(u32 tdim0, u32 tdim1,
                                      u32 tile0, u32 tile1, u64 stride0) {
    v8i g1;
    g1.s0 = (int)(2u << 16);                              // data_size = 4 bytes
    g1.s1 = (int)((tdim0 & 0xFFFFu) << 16);               // tensor_dim0[15:0]
    g1.s2 = (int)((tdim0 >> 16) | ((tdim1 & 0xFFFFu) << 16));
    g1.s3 = (int)((tdim1 >> 16) | (tile0 << 16));         // tile_dim0
    g1.s4 = (int)(tile1 & 0xFFFFu);                       // tile_dim1 (tile_dim2=0)
    g1.s5 = (int)(u32)(stride0 & 0xFFFFFFFFull);          // dim0_stride[31:0]
    g1.s6 = (int)(u32)((stride0 >> 32) & 0xFFFFu);        // dim0_stride[47:32]
    g1.s7 = 0;
    return g1;
}

__global__ __launch_bounds__(256)
void vq_scan_kernel(const float* __restrict__ z,
                    const float* __restrict__ cb,
                    float* __restrict__ out_q,        // d_out + 1, [B,D]
                    float* __restrict__ partial) {    // per-block loss sums
    __shared__ __align__(16) float lds_cb[DT * K_];   // 16 KB codebook tile
    __shared__ __align__(16) float lds_z [BT * DT];   // 1 KB  z tile [b][d]
    __shared__ __align__(16) float lds_q [BT * DT];   // 1 KB  q tile [b][d]
    __shared__ float wsum[DT];

    const u32 t    = threadIdx.x;
    const u32 wave = t >> 5;
    const u32 lane = t & 31u;
    const u32 b0   = blockIdx.x * BT;
    const u32 d0   = blockIdx.y * DT;

    // ---- Stage codebook tile [DT x K] and z tile [BT x DT] via TDM ----
    if (t < 32u) {
        v4i gz4 = {0, 0, 0, 0};                 // groups 2/3 unused (2D tensors)
        v8i gz8 = {0, 0, 0, 0, 0, 0, 0, 0};

        u64 cb_ga = (u64)(cb + (size_t)d0 * K_);
        v4u g0a = tdm_g0((u32)(u64)&lds_cb[0], cb_ga);
        v8i g1a = tdm_g1(K_, D_, K_, DT, (u64)K_);
        __builtin_amdgcn_tensor_load_to_lds(g0a, g1a, gz4, gz4, gz8, 0);

        u64 z_ga = (u64)(z + (size_t)b0 * D_ + d0);
        v4u g0b = tdm_g0((u32)(u64)&lds_z[0], z_ga);
        v8i g1b = tdm_g1(D_, B_, DT, BT, (u64)D_);
        __builtin_amdgcn_tensor_load_to_lds(g0b, g1b, gz4, gz4, gz8, 0);

        __builtin_amdgcn_s_wait_tensorcnt(0);
    }
    __syncthreads();

    // ---- Per-thread scan: wave w owns dim d0+w, lane l owns batch b0+l.
    //      All lanes of a wave read the SAME LDS word -> broadcast, no conflicts.
    const float zv = lds_z[lane * DT + wave];
    const float4* __restrict__ row = (const float4*)(lds_cb + wave * K_);

    float bestS = 3.402823466e+38f;
    float bestC = 0.0f;
    #pragma unroll 4
    for (int k = 0; k < K_ / 4; ++k) {
        float4 c = row[k];
        {   float df = c.x - zv; float s = df * df;
            bool p = s < bestS; bestS = p ? s : bestS; bestC = p ? c.x : bestC; }
        {   float df = c.y - zv; float s = df * df;
            bool p = s < bestS; bestS = p ? s : bestS; bestC = p ? c.y : bestC; }
        {   float df = c.z - zv; float s = df * df;
            bool p = s < bestS; bestS = p ? s : bestS; bestC = p ? c.z : bestC; }
        {   float df = c.w - zv; float s = df * df;
            bool p = s < bestS; bestS = p ? s : bestS; bestC = p ? c.w : bestC; }
    }

    // quantized value for this (b,d); stash in LDS tile for the TDM store
    lds_q[lane * DT + wave] = bestC;

    // ---- Loss partial: (q - z)^2, wave32 shuffle tree, then per-block sum ----
    float e = bestC - zv; e = e * e;
    #pragma unroll
    for (int off = 16; off > 0; off >>= 1)
        e += __shfl_xor(e, off, 32);
    if (lane == 0) wsum[wave] = e;
    __syncthreads();

    if (t == 0) {
        float s = 0.0f;
        #pragma unroll
        for (int i = 0; i < DT; ++i) s += wsum[i];
        partial[blockIdx.y * gridDim.x + blockIdx.x] = s;
    }

    // ---- Store quantized tile [BT x DT] back to global via TDM ----
    if (t < 32u) {
        v4i gz4 = {0, 0, 0, 0};
        v8i gz8 = {0, 0, 0, 0, 0, 0, 0, 0};
        u64 q_ga = (u64)(out_q + (size_t)b0 * D_ + d0);
        v4u g0c = tdm_g0((u32)(u64)&lds_q[0], q_ga);
        v8i g1c = tdm_g1(D_, B_, DT, BT, (u64)D_);
        __builtin_amdgcn_tensor_store_from_lds(g0c, g1c, gz4, gz4, gz8, 0);
        __builtin_amdgcn_s_wait_tensorcnt(0);
    }
}

// Deterministic reduction of the 1024 per-block partials -> mean loss.
// One wave (EXEC all-ones, as WMMA requires). Each V_WMMA_F32_16X16X4_F32
// computes D = A x ones + C: row-sums of a 16x4 chunk of partials,
// accumulated over 16 chunks in the f32 accumulator (fixed order).
// A layout (16x4 f32, 2 VGPRs): lanes 0-15 hold K=0,1; lanes 16-31 hold K=2,3.
__global__ __launch_bounds__(32)
void vq_loss_reduce_kernel(const float* __restrict__ partial,
                           float* __restrict__ loss) {
    const u32 lane = threadIdx.x & 31u;
    const u32 row  = lane & 15u;        // A-matrix row M
    const u32 kh   = lane >> 4;         // 0 -> K={0,1}, 1 -> K={2,3}

    v2f bones; bones.x = 1.0f; bones.y = 1.0f;   // B = 4x16 all-ones
    v8f acc = {0.f, 0.f, 0.f, 0.f, 0.f, 0.f, 0.f, 0.f};

    #pragma unroll
    for (int ch = 0; ch < 16; ++ch) {            // 16 chunks x 64 = 1024
        const float* p = partial + ch * 64 + row * 4 + kh * 2;
        v2f a; a.x = p[0]; a.y = p[1];
        acc = __builtin_amdgcn_wmma_f32_16x16x4_f32(
            false, a, false, bones, (short)0, acc, false, false);
    }

    // D[m][n] per lane: lane n holds M=0..7 (n<16) or M=8..15 (n>=16).
    // Column total = sum of this lane's 8 values + partner lane (xor 16).
    float e = acc.s0 + acc.s1 + acc.s2 + acc.s3
            + acc.s4 + acc.s5 + acc.s6 + acc.s7;
    e += __shfl_xor(e, 16, 32);
    if (lane == 0) loss[0] = e * (1.0f / (float)(B_ * D_));
}

extern "C" void kernel_launch(void* const* d_in, const int* in_sizes, int n_in,
                              void* d_out, int out_size, void* d_ws, size_t ws_size,
                              hipStream_t stream) {
    const float* z  = (const float*)d_in[0];   // [B, D] f32
    const float* cb = (const float*)d_in[1];   // [D, K] f32
    float* out      = (float*)d_out;           // [0]=loss, [1..]=quantized [B,D]
    float* partial  = (float*)d_ws;            // (B/BT)*(D/DT) = 1024 floats

    dim3 grid(B_ / BT, D_ / DT);               // 64 x 16 = 1024 blocks
    vq_scan_kernel<<<grid, 256, 0, stream>>>(z, cb, out + 1, partial);
    vq_loss_reduce_kernel<<<1, 32, 0, stream>>>(partial, out);
}